// Retrace_9156870275137
// MI455X (gfx1250) — compile-verified
//
#include <hip/hip_runtime.h>
#include <stdint.h>

#define NDIM   4096
#define GAMMA  0.99f
#define WPB    8            // waves per block
#define CHUNK  32           // timesteps per wave iteration (== wave32)
#define NCHUNK (NDIM / CHUNK)

// ---------------- CDNA5 async global->LDS helpers ----------------
__device__ __forceinline__ uint32_t lds_off_u32(const void* p) {
    // addrspace(3) aperture is 4GiB aligned on AMDGPU: low 32 bits of the
    // generic address are the LDS byte offset the async unit expects.
    return (uint32_t)(uintptr_t)p;
}

__device__ __forceinline__ void async_ld_b32(uint32_t lds_addr, uint32_t voff,
                                             const float* sbase) {
    // GVS mode: mem_addr = SGPR64 + VGPR_I32 ; LDS dst = per-lane VGPR addr
    asm volatile("global_load_async_to_lds_b32 %0, %1, %2"
                 :: "v"(lds_addr), "v"(voff), "s"(sbase)
                 : "memory");
}

__device__ __forceinline__ void wait_async_le6() {
    asm volatile("s_wait_asynccnt 6" ::: "memory");
}
__device__ __forceinline__ void wait_async_0() {
    asm volatile("s_wait_asynccnt 0" ::: "memory");
}

// array slots in the LDS chunk buffer
// 0: Q[t]  1: r[t]  2: tpp[t+1]  3: bpp[t+1]  4: tq[t+1]  5: eq[t+1]
__device__ __forceinline__ void issue_chunk(uint32_t lds_base, int tBase, int lane,
                                            const float* Qr,  const float* Rr,
                                            const float* TPr, const float* BPr,
                                            const float* TQr, const float* EQr) {
    int t  = tBase + lane;
    int t1 = t + 1; if (t1 > NDIM - 1) t1 = NDIM - 1;   // clamp (only top lane of top chunk)
    uint32_t vt  = (uint32_t)(t  * 4);
    uint32_t vt1 = (uint32_t)(t1 * 4);
    uint32_t l   = lds_base + (uint32_t)(lane * 4);
    async_ld_b32(l + 0 * (CHUNK * 4), vt,  Qr);
    async_ld_b32(l + 1 * (CHUNK * 4), vt,  Rr);
    async_ld_b32(l + 2 * (CHUNK * 4), vt1, TPr);
    async_ld_b32(l + 3 * (CHUNK * 4), vt1, BPr);
    async_ld_b32(l + 4 * (CHUNK * 4), vt1, TQr);
    async_ld_b32(l + 5 * (CHUNK * 4), vt1, EQr);
}

__global__ __launch_bounds__(WPB * 32) void retrace_kernel(
        const float* __restrict__ Q,   const float* __restrict__ eQ,
        const float* __restrict__ tQ,  const float* __restrict__ rw,
        const float* __restrict__ tpp, const float* __restrict__ bpp,
        float* __restrict__ out) {
    __shared__ float sbuf[3][WPB][6][CHUNK];   // triple-buffered chunk staging

    const int lane = threadIdx.x & 31;
    const int wv   = threadIdx.x >> 5;
    const int row  = blockIdx.x * WPB + wv;
    const size_t rb = (size_t)row * NDIM;

    const float* Qr  = Q   + rb;
    const float* Rr  = rw  + rb;
    const float* TPr = tpp + rb;
    const float* BPr = bpp + rb;
    const float* TQr = tQ  + rb;
    const float* EQr = eQ  + rb;

    // Prologue: two chunk-groups in flight (depth-2 pipeline, 3 LDS buffers)
    {
        uint32_t b1 = lds_off_u32(&sbuf[(NCHUNK - 1) % 3][wv][0][0]);
        issue_chunk(b1, (NCHUNK - 1) * CHUNK, lane, Qr, Rr, TPr, BPr, TQr, EQr);
        uint32_t b0 = lds_off_u32(&sbuf[(NCHUNK - 2) % 3][wv][0][0]);
        issue_chunk(b0, (NCHUNK - 2) * CHUNK, lane, Qr, Rr, TPr, BPr, TQr, EQr);
    }

    double lane_sum = 0.0;
    float  carry    = 0.0f;   // q_{t+32} of the chunk above; unused for top chunk (A==0 chain)

    for (int c = NCHUNK - 1; c >= 0; --c) {
        // async loads retire in order: <=6 outstanding => group for chunk c is done
        if (c >= 1) wait_async_le6(); else wait_async_0();

        // keep pipeline full
        if (c >= 2) {
            uint32_t bn = lds_off_u32(&sbuf[(c - 2) % 3][wv][0][0]);
            issue_chunk(bn, (c - 2) * CHUNK, lane, Qr, Rr, TPr, BPr, TQr, EQr);
        }

        const int cur = c % 3;
        const int t   = c * CHUNK + lane;

        float q_in = sbuf[cur][wv][0][lane];
        float r    = sbuf[cur][wv][1][lane];
        float dpp  = sbuf[cur][wv][2][lane] - sbuf[cur][wv][3][lane];
        float tq1  = sbuf[cur][wv][4][lane];
        float eq1  = sbuf[cur][wv][5][lane];

        float cc = __expf(fminf(dpp, 0.0f));
        float A  = GAMMA * cc;                                  // a_t
        float B  = fmaf(-A, tq1, fmaf(GAMMA, eq1, 100.0f * r)); // b_t
        if (t == NDIM - 1) { A = 0.0f; B = tq1; }               // q_{T-1} = tq[T-1]

        // suffix composition scan across lanes: (A_j,B_j) = f_j o ... o f_31
        #pragma unroll
        for (int d = 1; d < 32; d <<= 1) {
            float Au = __shfl_down(A, d, 32);
            float Bu = __shfl_down(B, d, 32);
            bool ok = (lane + d) < 32;
            Au = ok ? Au : 1.0f;
            Bu = ok ? Bu : 0.0f;
            B = fmaf(A, Bu, B);
            A = A * Au;
        }

        float q = fmaf(A, carry, B);      // Q_ret[row][t]
        float e = q_in - q;
        lane_sum += (double)(e * e);

        carry = __shfl(q, 0, 32);         // q at lowest t of this chunk
    }

    // wave reduction, then one atomic per wave
    #pragma unroll
    for (int d = 16; d >= 1; d >>= 1) lane_sum += __shfl_down(lane_sum, d, 32);
    if (lane == 0) {
        const double inv = 1.0 / ((double)NDIM * (double)NDIM);
        atomicAdd(out, (float)(lane_sum * inv));
    }
}

__global__ void retrace_zero(float* out) {
    if (threadIdx.x == 0) out[0] = 0.0f;
}

extern "C" void kernel_launch(void* const* d_in, const int* in_sizes, int n_in,
                              void* d_out, int out_size, void* d_ws, size_t ws_size,
                              hipStream_t stream) {
    (void)in_sizes; (void)n_in; (void)out_size; (void)d_ws; (void)ws_size;
    const float* Q   = (const float*)d_in[0];
    const float* eQ  = (const float*)d_in[1];
    const float* tQ  = (const float*)d_in[2];
    const float* rw  = (const float*)d_in[3];
    const float* tpp = (const float*)d_in[4];
    const float* bpp = (const float*)d_in[5];
    float* out = (float*)d_out;

    retrace_zero<<<1, 32, 0, stream>>>(out);
    retrace_kernel<<<NDIM / WPB, WPB * 32, 0, stream>>>(Q, eQ, tQ, rw, tpp, bpp, out);
}